// Connection_gnew_17076789969604
// MI455X (gfx1250) — compile-verified
//
#include <hip/hip_runtime.h>
#include <math.h>

typedef float v2f __attribute__((ext_vector_type(2)));
typedef float v8f __attribute__((ext_vector_type(8)));

#define D  64
#define QS 65   // padded LDS stride for q   (bank-conflict-free column reads)
#define SS 65   // padded LDS stride for S
#define GS 66   // padded LDS stride for augmented [g | P]

// One workgroup per sample. 128 threads = 4 wave32s.
__global__ __launch_bounds__(128) void connection_gnew_kernel(
    const float* __restrict__ x,  const float* __restrict__ v,
    const float* __restrict__ W1, const float* __restrict__ b1,
    const float* __restrict__ W2, const float* __restrict__ b2,
    const float* __restrict__ W3, float* __restrict__ out)
{
  __shared__ float sh_x[D], sh_v[D], sh_h[D], sh_y[D];
  __shared__ float sh_q[D * QS];   // q[m][l] = s[m]*W1[m][l]
  __shared__ float sh_g[D * GS];   // augmented system [ g | P ]
  __shared__ float sh_S[D * SS];   // current S_i tile

  const int n    = blockIdx.x;
  const int tid  = threadIdx.x;
  const int lane = tid & 31;
  const int wv   = tid >> 5;       // wave id 0..3 -> owns rows 16*wv..16*wv+15
  const int lo   = lane & 15;
  const int hi   = lane >> 4;      // upper half-wave flag

  // ---- stage x, v ----
  if (tid < D) {
    sh_x[tid] = x[(size_t)n * D + tid];
    sh_v[tid] = v[(size_t)n * D + tid];
  }
  __syncthreads();

  // ---- h = tanh(x@W1^T + b1); s = 1-h^2; q = diag(s)@W1 ----
  if (tid < D) {
    float acc = b1[tid];
    for (int l = 0; l < D; ++l) acc += sh_x[l] * W1[tid * D + l];
    float hm = tanhf(acc);
    sh_h[tid] = hm;
    float sm = 1.0f - hm * hm;
    for (int c = 0; c < D; ++c) sh_q[tid * QS + c] = sm * W1[tid * D + c];
  }
  __syncthreads();

  // ---- g[i][j] = sum_m h_m (W2[i*64+j,m]+W2[j*64+i,m]) + b2[i*64+j]+b2[j*64+i]
  for (int e = tid; e < D * D; e += 128) {
    int i = e >> 6, j = e & 63;
    const float* r0 = W2 + (size_t)(i * D + j) * D;
    const float* r1 = W2 + (size_t)(j * D + i) * D;
    float acc = b2[i * D + j] + b2[j * D + i];
    for (int m = 0; m < D; ++m) acc += sh_h[m] * (r0[m] + r1[m]);
    sh_g[i * GS + j] = acc;
  }
  __syncthreads();

  // ---- main loop: S_i = Asum_i @ q via fp32 WMMA; accumulate P ----
  float Preg = 0.0f;
  const int jrow = 16 * wv + lo;   // this lane's A-matrix row (M index)

  for (int i = 0; i < D; ++i) {
    // A fragments (16x64 strip, symmetrized W2 rows), ISA 16x4 f32 A layout:
    // lanes 0-15 hold K={4kc,4kc+1}, lanes 16-31 hold K={4kc+2,4kc+3}
    v2f afrag[16];
    {
      const float* ra = W2 + (size_t)(i * D + jrow) * D;
      const float* rb = W2 + (size_t)(jrow * D + i) * D;
      #pragma unroll
      for (int kc = 0; kc < 16; ++kc) {
        int k0 = 4 * kc + 2 * hi;
        afrag[kc].x = ra[k0]     + rb[k0];
        afrag[kc].y = ra[k0 + 1] + rb[k0 + 1];
      }
    }
    // 4 column tiles of the 16x64 output strip
    #pragma unroll
    for (int tc = 0; tc < 4; ++tc) {
      v8f acc = {0.f, 0.f, 0.f, 0.f, 0.f, 0.f, 0.f, 0.f};
      #pragma unroll
      for (int kc = 0; kc < 16; ++kc) {
        int k0 = 4 * kc + 2 * hi;
        v2f b;
        b.x = sh_q[k0 * QS + 16 * tc + lo];
        b.y = sh_q[(k0 + 1) * QS + 16 * tc + lo];
        acc = __builtin_amdgcn_wmma_f32_16x16x4_f32(
            false, afrag[kc], false, b, (short)0, acc, false, false);
      }
      // C/D layout: VGPR r -> row (r + 8*hi), col lo within the tile
      #pragma unroll
      for (int r = 0; r < 8; ++r)
        sh_S[(16 * wv + r + 8 * hi) * SS + 16 * tc + lo] = acc[r];
    }
    __syncthreads();

    // P_l += v_i * sum_j v_j * S_i[j,l]^2 * S_i[l,j]   (thread t owns l=t)
    if (tid < D) {
      float a = 0.0f;
      for (int j = 0; j < D; ++j) {
        float sjl = sh_S[j * SS + tid];
        a += sh_v[j] * sjl * sjl * sh_S[tid * SS + j];
      }
      Preg += sh_v[i] * a;
    }
    __syncthreads();   // S reused next iteration
  }

  if (tid < D) sh_g[tid * GS + D] = Preg;   // RHS column
  __syncthreads();

  // ---- solve g*y = P (no pivoting: g ~ 6*I + small) ----
  for (int p = 0; p < D; ++p) {
    if (tid < D && tid > p) {
      float f = sh_g[tid * GS + p] / sh_g[p * GS + p];
      for (int c = p; c <= D; ++c)
        sh_g[tid * GS + c] -= f * sh_g[p * GS + c];
    }
    __syncthreads();
  }
  if (tid == 0) {
    for (int p = D - 1; p >= 0; --p) {
      float acc = sh_g[p * GS + D];
      for (int c = p + 1; c < D; ++c) acc -= sh_g[p * GS + c] * sh_y[c];
      sh_y[p] = acc / sh_g[p * GS + p];
    }
  }
  __syncthreads();

  // ---- dv = 0.5*y ; out = concat(v, dv) @ W3^T ----
  if (tid < D) {
    const float* w3row = W3 + (size_t)tid * (2 * D);
    float acc = 0.0f;
    for (int c = 0; c < D; ++c) acc += sh_v[c] * w3row[c];
    for (int c = 0; c < D; ++c) acc += 0.5f * sh_y[c] * w3row[D + c];
    out[(size_t)n * D + tid] = acc;
  }
}

extern "C" void kernel_launch(void* const* d_in, const int* in_sizes, int n_in,
                              void* d_out, int out_size, void* d_ws, size_t ws_size,
                              hipStream_t stream) {
  // setup_inputs order: t, x, v, W1, b1, W2, b2, W3 (t unused)
  const float* x  = (const float*)d_in[1];
  const float* v  = (const float*)d_in[2];
  const float* W1 = (const float*)d_in[3];
  const float* b1 = (const float*)d_in[4];
  const float* W2 = (const float*)d_in[5];
  const float* b2 = (const float*)d_in[6];
  const float* W3 = (const float*)d_in[7];
  float* out = (float*)d_out;

  const int n = in_sizes[1] / D;   // 256 samples
  connection_gnew_kernel<<<n, 128, 0, stream>>>(x, v, W1, b1, W2, b2, W3, out);
}